// DNN_SKalmanNet_GSS_21036749816494
// MI455X (gfx1250) — compile-verified
//
#include <hip/hip_runtime.h>
#include <hip/hip_bf16.h>
#include <math.h>

// ---------------------------------------------------------------------------
// DNN_SKalmanNet_GSS on gfx1250 (MI455X).
// Bandwidth-bound fp32 matvec chain (~498 MB of weights per call -> ~21us
// floor at 23.3 TB/s).  Matvecs run on the WMMA pipe via
// V_WMMA_F32_16X16X4_F32 with the x-vector broadcast across B's 16 columns,
// so each wave produces y for a 16-row tile of W while streaming W at 1x.
// ---------------------------------------------------------------------------

typedef __attribute__((ext_vector_type(2))) float v2f;
typedef __attribute__((ext_vector_type(8))) float v8f;

#define XD   32
#define YD   32
#define H1   5120      // (X+Y)*10*8
#define H2   4096      // X*Y*4
#define HID  2048      // X*X + Y*Y
#define IN1  1120      // 2X + Y + X*Y
#define G3   6144      // 3*HID
#define OUT1 1024      // X*X

// ---------------------------------------------------------------------------
// y[m] = act( sum_k W[m][k]*x[k] + bias[m] ),  one 16-row tile per wave.
//
// WMMA f32 16x16x4 operand mapping (wave32):
//   A (16x4):  lanes 0-15 -> M=lane, VGPR{0,1} = K{0,1};
//              lanes 16-31 -> M=lane-16, VGPR{0,1} = K{2,3}.
//   B (4x16):  x broadcast into every column N, same K split by lane half.
//   D (16x16): VGPR v: lanes 0-15 -> M=v, lanes 16-31 -> M=v+8; since every
//              column of D is identical, lane 0 / lane 16 hold the 16 sums.
// ---------------------------------------------------------------------------
__global__ __launch_bounds__(256)
void matvec16_wmma(const float* __restrict__ W, const float* __restrict__ x,
                   const float* __restrict__ bias, float* __restrict__ y,
                   int M, int K, int relu_act)
{
    const int wave = (int)((blockIdx.x * blockDim.x + threadIdx.x) >> 5);
    const int lane = (int)(threadIdx.x & 31);
    const int row0 = wave * 16;
    if (row0 >= M) return;                 // wave-uniform: EXEC stays all-1s

    const int r    = lane & 15;            // row within tile
    const int koff = (lane >> 4) << 1;     // 0 for lanes 0-15, 2 for 16-31
    const float* __restrict__ wrow =
        W + (size_t)(row0 + r) * (size_t)K + (size_t)koff;

    v8f acc = {};
#pragma unroll 4
    for (int kb = 0; kb < K; kb += 4) {
        v2f a = *(const v2f*)(wrow + kb);      // 2 consecutive weights / lane
        v2f b;
        b.x = x[kb + koff];                    // broadcast x into B columns
        b.y = x[kb + koff + 1];
        acc = __builtin_amdgcn_wmma_f32_16x16x4_f32(
            /*neg_a=*/false, a, /*neg_b=*/false, b,
            /*c_mod=*/(short)0, acc, /*reuse_a=*/false, /*reuse_b=*/false);
    }

    // Lane 0 holds y[row0+0..7], lane 16 holds y[row0+8..15].
    if (r == 0) {
        const int mbase = row0 + ((lane >> 4) << 3);
#pragma unroll
        for (int v = 0; v < 8; ++v) {
            float val = acc[v] + bias[mbase + v];
            if (relu_act) val = fmaxf(val, 0.0f);
            y[mbase + v] = val;
        }
    }
}

// Build input1 = [state_inno, diff_state, lin_err, Jacobian]
//       input2 = [obs_inno,   diff_obs,   lin_err, Jacobian]
__global__ void prep_inputs(const float* __restrict__ s,  const float* __restrict__ o,
                            const float* __restrict__ ds, const float* __restrict__ dob,
                            const float* __restrict__ le, const float* __restrict__ J,
                            float* __restrict__ in1, float* __restrict__ in2)
{
    int i = (int)(blockIdx.x * blockDim.x + threadIdx.x);
    if (i >= IN1) return;
    float a, b;
    if (i < 32)       { a = s[i];        b = o[i]; }
    else if (i < 64)  { a = ds[i - 32];  b = dob[i - 32]; }
    else if (i < 96)  { a = le[i - 64];  b = le[i - 64]; }
    else              { a = J[i - 96];   b = J[i - 96]; }
    in1[i] = a;
    in2[i] = b;
}

// PyTorch GRU cell combine (gate order r,z,n); gi/gh already include biases.
__global__ void gru_combine(const float* __restrict__ gi, const float* __restrict__ gh,
                            const float* __restrict__ h,  float* __restrict__ hout)
{
    int i = (int)(blockIdx.x * blockDim.x + threadIdx.x);
    if (i >= HID) return;
    float r = 1.0f / (1.0f + __expf(-(gi[i]           + gh[i])));
    float z = 1.0f / (1.0f + __expf(-(gi[HID + i]     + gh[HID + i])));
    float n = tanhf(gi[2 * HID + i] + r * gh[2 * HID + i]);
    hout[i] = (1.0f - z) * n + z * h[i];
}

extern "C" void kernel_launch(void* const* d_in, const int* in_sizes, int n_in,
                              void* d_out, int out_size, void* d_ws, size_t ws_size,
                              hipStream_t stream)
{
    (void)in_sizes; (void)n_in; (void)out_size; (void)ws_size;

    const float* state_inno = (const float*)d_in[0];
    const float* obs_inno   = (const float*)d_in[1];
    const float* diff_state = (const float*)d_in[2];
    const float* diff_obs   = (const float*)d_in[3];
    const float* lin_err    = (const float*)d_in[4];
    const float* Jac        = (const float*)d_in[5];
    const float* l1_W  = (const float*)d_in[6];
    const float* l1_b  = (const float*)d_in[7];
    const float* g1Wih = (const float*)d_in[8];
    const float* g1Whh = (const float*)d_in[9];
    const float* g1bih = (const float*)d_in[10];
    const float* g1bhh = (const float*)d_in[11];
    const float* l2_W1 = (const float*)d_in[12];
    const float* l2_b1 = (const float*)d_in[13];
    const float* l2_W2 = (const float*)d_in[14];
    const float* l2_b2 = (const float*)d_in[15];
    const float* l3_W  = (const float*)d_in[16];
    const float* l3_b  = (const float*)d_in[17];
    const float* g2Wih = (const float*)d_in[18];
    const float* g2Whh = (const float*)d_in[19];
    const float* g2bih = (const float*)d_in[20];
    const float* g2bhh = (const float*)d_in[21];
    const float* l4_W1 = (const float*)d_in[22];
    const float* l4_b1 = (const float*)d_in[23];
    const float* l4_W2 = (const float*)d_in[24];
    const float* l4_b2 = (const float*)d_in[25];
    const float* hn1   = (const float*)d_in[26];
    const float* hn2   = (const float*)d_in[27];

    float* out = (float*)d_out;          // Pk at [0,1024), Sk at [1024,2048)

    // Workspace layout (floats). Branch-2 reuses gi/gh/t (stream-ordered).
    float* ws  = (float*)d_ws;
    float* in1 = ws;                     // 1120
    float* in2 = ws + 1120;              // 1120
    float* lo  = ws + 2240;              // 5120  (l1/l3 output, reused)
    float* gi  = ws + 7360;              // 6144
    float* gh  = ws + 13504;             // 6144
    float* hx  = ws + 19648;             // 2048  (new hidden state, reused)
    float* t   = ws + 21696;             // 4096  (l2/l4 hidden, reused)

    const dim3 blk(256);
    auto mv = [&](const float* W, const float* x, const float* b, float* y,
                  int M, int K, int act) {
        matvec16_wmma<<<dim3((unsigned)(M / 128)), blk, 0, stream>>>(W, x, b, y, M, K, act);
    };

    prep_inputs<<<dim3((IN1 + 255) / 256), blk, 0, stream>>>(
        state_inno, obs_inno, diff_state, diff_obs, lin_err, Jac, in1, in2);

    // ---- Branch 1: Pk ----
    mv(l1_W,  in1, l1_b,  lo, H1,  IN1, 1);        // relu(l1_W @ input1 + b)
    mv(g1Wih, lo,  g1bih, gi, G3,  H1,  0);        // gi = Wih @ l1_out + bih
    mv(g1Whh, hn1, g1bhh, gh, G3,  HID, 0);        // gh = Whh @ hn1   + bhh
    gru_combine<<<dim3(HID / 256), blk, 0, stream>>>(gi, gh, hn1, hx);
    mv(l2_W1, hx, l2_b1, t,   H2,   HID, 1);       // relu(l2_W1 @ h + b1)
    mv(l2_W2, t,  l2_b2, out, OUT1, H2,  0);       // Pk

    // ---- Branch 2: Sk ----
    mv(l3_W,  in2, l3_b,  lo, H1,  IN1, 1);
    mv(g2Wih, lo,  g2bih, gi, G3,  H1,  0);
    mv(g2Whh, hn2, g2bhh, gh, G3,  HID, 0);
    gru_combine<<<dim3(HID / 256), blk, 0, stream>>>(gi, gh, hn2, hx);
    mv(l4_W1, hx, l4_b1, t,          H2,   HID, 1);
    mv(l4_W2, t,  l4_b2, out + OUT1, OUT1, H2,  0); // Sk
}